// StressCapsuleLayer_10015863734392
// MI455X (gfx1250) — compile-verified
//
#include <hip/hip_runtime.h>
#include <math.h>

// Capsule routing, fused recompute strategy for MI455X (gfx1250).
// u_hat (340 MB fp32) is never materialized: each consumer recomputes the
// 16x1024 per-(b_tile,i) tile with V_WMMA_F32_16X16X4_F32 from x (5.3 MB) and
// W (42.5 MB, L2-resident). 5 WMMA passes total (3 accum + 2 agreement).

#define B_      128
#define N_IN    648
#define D_IN    16
#define N_OUT   32
#define D_OUT   32
#define NCOL    1024          // N_OUT * D_OUT
#define EPSV    1e-7f
#define ICHUNKS 18
#define ILEN    36            // 648 / 18

typedef float v2f __attribute__((ext_vector_type(2)));
typedef float v8f __attribute__((ext_vector_type(8)));

__device__ __forceinline__ v8f wmma_16x16x4_f32(v2f a, v2f b, v8f c) {
  // 8 args: (neg_a, A, neg_b, B, c_mod, C, reuse_a, reuse_b)
  return __builtin_amdgcn_wmma_f32_16x16x4_f32(false, a, false, b, (short)0, c,
                                               false, false);
}

__global__ void zero_kernel(float* __restrict__ p, int n) {
  int i = blockIdx.x * blockDim.x + threadIdx.x;
  if (i < n) p[i] = 0.0f;
}

// softmax over j (N_OUT=32) per (b, i) row; logits -> c
__global__ void softmax_kernel(const float* __restrict__ logits,
                               float* __restrict__ c) {
  int row = blockIdx.x * blockDim.x + threadIdx.x;
  if (row >= B_ * N_IN) return;
  const float* l = logits + (size_t)row * N_OUT;
  float*       o = c      + (size_t)row * N_OUT;
  float m = l[0];
  #pragma unroll
  for (int j = 1; j < N_OUT; ++j) m = fmaxf(m, l[j]);
  float e[N_OUT];
  float sum = 0.0f;
  #pragma unroll
  for (int j = 0; j < N_OUT; ++j) { e[j] = expf(l[j] - m); sum += e[j]; }
  float inv = 1.0f / sum;
  #pragma unroll
  for (int j = 0; j < N_OUT; ++j) o[j] = e[j] * inv;
}

// s[b, j, d] += sum_{i in chunk} c[b,i,j] * u_hat[b,i,j,d]
// grid: (ICHUNKS, 8 n-splits of 128 cols, 8 b-tiles of 16), block = 1 wave.
__global__ void __launch_bounds__(32)
accum_kernel(const float* __restrict__ x,
             const float* __restrict__ W,
             const float* __restrict__ c,
             float* __restrict__ s) {
  const int lane = threadIdx.x;
  const int half = lane >> 4;   // K/M half-select per WMMA layout
  const int l16  = lane & 15;
  const int i0     = blockIdx.x * ILEN;
  const int n_base = blockIdx.y * 128;
  const int b_base = blockIdx.z * 16;

  const v8f zero = {};
  v8f sacc[8];
  #pragma unroll
  for (int t = 0; t < 8; ++t) sacc[t] = zero;

  for (int ii = 0; ii < ILEN; ++ii) {
    const int i = i0 + ii;
    // A tile: x[b_base..+15, i, 0..15]; VGPR0=K0/K2, VGPR1=K1/K3 by lane half
    const float* xrow = x + ((size_t)(b_base + l16) * N_IN + i) * D_IN;
    v2f A[4];
    #pragma unroll
    for (int ks = 0; ks < 4; ++ks) {
      A[ks].x = xrow[ks * 4 + 2 * half];
      A[ks].y = xrow[ks * 4 + 1 + 2 * half];
    }
    #pragma unroll
    for (int t = 0; t < 8; ++t) {
      const int ncol = n_base + t * 16;       // 16-wide tile inside one j
      const int j    = ncol >> 5;
      const int d    = (ncol & 31) + l16;     // column within capsule j
      const float* wb = W + (((size_t)i * N_OUT + j) * D_OUT + d) * D_IN;
      // Batch all 4 K-step B operands before the MAC chain so one
      // s_wait_loadcnt covers the round trip instead of four.
      v2f Bv[4];
      #pragma unroll
      for (int ks = 0; ks < 4; ++ks) {
        Bv[ks].x = wb[ks * 4 + 2 * half];
        Bv[ks].y = wb[ks * 4 + 1 + 2 * half];
      }
      v8f U = wmma_16x16x4_f32(A[0], Bv[0], zero);
      U     = wmma_16x16x4_f32(A[1], Bv[1], U);
      U     = wmma_16x16x4_f32(A[2], Bv[2], U);
      U     = wmma_16x16x4_f32(A[3], Bv[3], U);
      #pragma unroll
      for (int r = 0; r < 8; ++r) {           // C row = r + 8*half
        const int b = b_base + r + 8 * half;
        const float cv = c[((size_t)b * N_IN + i) * N_OUT + j];
        sacc[t][r] += cv * U[r];
      }
    }
  }
  #pragma unroll
  for (int t = 0; t < 8; ++t) {
    #pragma unroll
    for (int r = 0; r < 8; ++r) {
      const int b = b_base + r + 8 * half;
      atomicAdd(&s[(size_t)b * NCOL + n_base + t * 16 + l16], sacc[t][r]);
    }
  }
}

// logits[b,i,j] += sum_d u_hat[b,i,j,d] * v[b,j,d]
// grid: (N_IN, 8 n-splits, 8 b-tiles), block = 1 wave.
__global__ void __launch_bounds__(32)
agree_kernel(const float* __restrict__ x,
             const float* __restrict__ W,
             const float* __restrict__ v,
             float* __restrict__ logits) {
  const int lane = threadIdx.x;
  const int half = lane >> 4;
  const int l16  = lane & 15;
  const int i      = blockIdx.x;
  const int n_base = blockIdx.y * 128;
  const int b_base = blockIdx.z * 16;

  const float* xrow = x + ((size_t)(b_base + l16) * N_IN + i) * D_IN;
  v2f A[4];
  #pragma unroll
  for (int ks = 0; ks < 4; ++ks) {
    A[ks].x = xrow[ks * 4 + 2 * half];
    A[ks].y = xrow[ks * 4 + 1 + 2 * half];
  }
  const v8f zero = {};

  for (int u = 0; u < 4; ++u) {               // 4 capsules j per wave
    const int j = (n_base >> 5) + u;
    float agg[8];
    #pragma unroll
    for (int r = 0; r < 8; ++r) agg[r] = 0.0f;

    #pragma unroll
    for (int tp = 0; tp < 2; ++tp) {          // two 16-col tiles cover j
      const int dcol = tp * 16 + l16;
      const float* wb = W + (((size_t)i * N_OUT + j) * D_OUT + dcol) * D_IN;
      v2f Bv[4];
      #pragma unroll
      for (int ks = 0; ks < 4; ++ks) {
        Bv[ks].x = wb[ks * 4 + 2 * half];
        Bv[ks].y = wb[ks * 4 + 1 + 2 * half];
      }
      v8f U = wmma_16x16x4_f32(A[0], Bv[0], zero);
      U     = wmma_16x16x4_f32(A[1], Bv[1], U);
      U     = wmma_16x16x4_f32(A[2], Bv[2], U);
      U     = wmma_16x16x4_f32(A[3], Bv[3], U);
      #pragma unroll
      for (int r = 0; r < 8; ++r) {
        const int b = b_base + r + 8 * half;
        float p = U[r] * v[(size_t)b * NCOL + j * D_OUT + dcol];
        // reduce over 16 columns (lanes within each half)
        p += __shfl_xor(p, 1);
        p += __shfl_xor(p, 2);
        p += __shfl_xor(p, 4);
        p += __shfl_xor(p, 8);
        agg[r] += p;
      }
    }
    if (l16 == 0) {                           // lane 0 and lane 16 write
      #pragma unroll
      for (int r = 0; r < 8; ++r) {
        const int b = b_base + r + 8 * half;
        const size_t idx = ((size_t)b * N_IN + i) * N_OUT + j;
        logits[idx] += agg[r];
      }
    }
  }
}

// v = squash(s) per (b, j) over D_OUT=32; one wave per (b, j)
__global__ void __launch_bounds__(32)
squash_kernel(const float* __restrict__ s, float* __restrict__ out) {
  const int d  = threadIdx.x;                 // 0..31
  const int bj = blockIdx.x;                  // b*N_OUT + j
  const size_t idx = (size_t)bj * D_OUT + d;
  const float sv = s[idx];
  float p = sv * sv;
  p += __shfl_xor(p, 1);
  p += __shfl_xor(p, 2);
  p += __shfl_xor(p, 4);
  p += __shfl_xor(p, 8);
  p += __shfl_xor(p, 16);
  const float scale = p / ((1.0f + p) * sqrtf(p + EPSV));
  out[idx] = scale * sv;
}

extern "C" void kernel_launch(void* const* d_in, const int* in_sizes, int n_in,
                              void* d_out, int out_size, void* d_ws,
                              size_t ws_size, hipStream_t stream) {
  const float* x = (const float*)d_in[0];   // [128, 648, 16]
  const float* W = (const float*)d_in[1];   // [648, 32, 32, 16]
  float* out = (float*)d_out;               // [128, 32, 32]
  float* ws  = (float*)d_ws;

  const size_t Lg = (size_t)B_ * N_IN * N_OUT;   // 2,654,208 floats
  float* logits = ws;
  float* cbuf   = ws + Lg;
  float* sbuf   = ws + 2 * Lg;
  float* vbuf   = sbuf + (size_t)B_ * NCOL;

  const dim3 wave(32);
  const dim3 accGrid(ICHUNKS, 8, 8);
  const dim3 agrGrid(N_IN, 8, 8);
  const int nrow = B_ * N_IN;
  const int nS   = B_ * NCOL;

  zero_kernel<<<((int)Lg + 255) / 256, 256, 0, stream>>>(logits, (int)Lg);
  for (int it = 0; it < 3; ++it) {
    if (it > 0)
      agree_kernel<<<agrGrid, wave, 0, stream>>>(x, W, vbuf, logits);
    softmax_kernel<<<(nrow + 255) / 256, 256, 0, stream>>>(logits, cbuf);
    zero_kernel<<<(nS + 255) / 256, 256, 0, stream>>>(sbuf, nS);
    accum_kernel<<<accGrid, wave, 0, stream>>>(x, W, cbuf, sbuf);
    squash_kernel<<<B_ * N_OUT, 32, 0, stream>>>(sbuf, (it == 2) ? out : vbuf);
  }
}